// clustering_attention_dynamic_learning2_45286135169495
// MI455X (gfx1250) — compile-verified
//
#include <hip/hip_runtime.h>

// ---------------------------------------------------------------------------
// Problem constants (from the reference)
// ---------------------------------------------------------------------------
#define BB 32
#define NN 207
#define SS 12          // S
#define SOO 12         // SO
#define CC 10          // C
#define KK 20          // K
#define MM (BB * NN)   // 6624 rows, = 414 * 16 exactly
#define PW 16          // padded column stride for wh/ax/ay
#define NEG_SLOPE 0.01f
#define NTILES (MM / 16)   // 414
#define TPW 3              // row-tiles per wave in the WMMA kernel

// Workspace layout (in floats)
#define WS_ACC  0                       // 8 accumulators: wh_sum, att_sum, dist_sum, cl_sum, ccl_sum
#define WS_BW   16                      // 16x12 zero-padded  W  as B-matrix   (B[k][n] = W[n][k])
#define WS_B1   256                     // 16x12 zero-padded (a1 W) as B-matrix
#define WS_B2   512                     // 16x12 zero-padded (a2 W) as B-matrix
#define WS_WH   768                     // MM x 16 (cols 12..15 are exact zeros)
#define WS_AX   (WS_WH + MM * PW)       // MM x 16 (cols 10..15 zeros)
#define WS_AY   (WS_AX + MM * PW)       // MM x 16 (cols 10..15 zeros)
#define WS_ATT  (WS_AY + MM * PW)       // MM x K x C  pre-softmax attention

typedef float v2f __attribute__((ext_vector_type(2)));
typedef float v8f __attribute__((ext_vector_type(8)));

// ---------------------------------------------------------------------------
// Kernel 0: zero accumulators; build zero-padded B operands:
//   BW[n][k] = W[n][k]          (n<12)
//   B1[n][k] = (a1 W)[n][k]     (n<10)
//   B2[n][k] = (a2 W)[n][k]     (n<10)
// ---------------------------------------------------------------------------
__global__ void k_setup(const float* __restrict__ aw, const float* __restrict__ ww,
                        float* __restrict__ ws) {
  __shared__ float s1[CC * SS], s2[CC * SS];
  const int tid = threadIdx.x;
  if (tid < 8) ws[WS_ACC + tid] = 0.0f;
  if (tid < CC * SS) {
    const int c = tid / SS, s = tid % SS;
    float a = 0.0f, b = 0.0f;
    for (int j = 0; j < SOO; ++j) {
      const float w = ww[j * SS + s];
      a += aw[c * 2 * SOO + j] * w;         // a1[c][j]
      b += aw[c * 2 * SOO + SOO + j] * w;   // a2[c][j]
    }
    s1[tid] = a; s2[tid] = b;
  }
  __syncthreads();
  if (tid < 16 * SS) {
    const int n = tid / SS, k = tid % SS;
    ws[WS_BW + tid] = (n < SOO) ? ww[n * SS + k] : 0.0f;
    ws[WS_B1 + tid] = (n < CC)  ? s1[n * SS + k] : 0.0f;
    ws[WS_B2 + tid] = (n < CC)  ? s2[n * SS + k] : 0.0f;
  }
}

// ---------------------------------------------------------------------------
// Kernel 1: fused wh / ax / ay via V_WMMA_F32_16X16X4_F32.
// Each wave: load 9 B chunk registers once, then 3 row-tiles x 9 WMMAs.
// All loads/stores unconditional thanks to zero padding.
// A layout (16x4 f32): lane l holds M=l%16; VGPR0=K{k0|k0+2}, VGPR1=K+1.
// D layout: VGPR r -> M=r (lanes 0-15) / M=8+r (lanes 16-31), N=lane%16.
// ---------------------------------------------------------------------------
__global__ void k_wh_wmma(const float* __restrict__ x, float* __restrict__ ws) {
  const int lane = threadIdx.x & 31;
  const int wave = threadIdx.x >> 5;
  const int w    = blockIdx.x * 8 + wave;
  const int t0   = w * TPW;
  if (t0 >= NTILES) return;                 // uniform per wave
  const int n  = lane & 15;
  const int hi = lane >> 4;

  v2f bw[3], b1[3], b2[3];
  for (int c = 0; c < 3; ++c) {
    const int k0 = c * 4 + (hi << 1);
    bw[c] = *(const v2f*)(ws + WS_BW + n * SS + k0);
    b1[c] = *(const v2f*)(ws + WS_B1 + n * SS + k0);
    b2[c] = *(const v2f*)(ws + WS_B2 + n * SS + k0);
  }

  float whPartial = 0.0f;
  const int tEnd = (t0 + TPW < NTILES) ? t0 + TPW : NTILES;
  for (int t = t0; t < tEnd; ++t) {
    const int base = t * 16;
    const int row  = base + n;
    v8f cW = {}; v8f cX = {}; v8f cY = {};
    for (int c = 0; c < 3; ++c) {
      const int k0 = c * 4 + (hi << 1);
      v2f A = *(const v2f*)(x + row * SS + k0);
      cW = __builtin_amdgcn_wmma_f32_16x16x4_f32(false, A, false, bw[c], (short)0, cW, false, false);
      cX = __builtin_amdgcn_wmma_f32_16x16x4_f32(false, A, false, b1[c], (short)0, cX, false, false);
      cY = __builtin_amdgcn_wmma_f32_16x16x4_f32(false, A, false, b2[c], (short)0, cY, false, false);
    }
    for (int r = 0; r < 8; ++r) {
      const int m = base + hi * 8 + r;
      ws[WS_WH + m * PW + n] = cW[r];
      ws[WS_AX + m * PW + n] = cX[r];
      ws[WS_AY + m * PW + n] = cY[r];
      whPartial += (n < SOO) ? cW[r] : 0.0f;
    }
  }
  for (int off = 16; off > 0; off >>= 1) whPartial += __shfl_down(whPartial, off);
  if (lane == 0) atomicAdd(&ws[WS_ACC + 0], whPartial);
}

// ---------------------------------------------------------------------------
// Kernel 2: per (b,i): attention gather + softmax over K, then the einsum
//   output[c,s] = sum_k soft[k,c]*wh_topk[k,s]   as a 16x16x20 WMMA (wave 0),
// plus cluster_center_loss partial and attention sum.
// ---------------------------------------------------------------------------
__global__ void k_attn(const int* __restrict__ adj, float* __restrict__ ws,
                       float* __restrict__ out) {
  const int bi  = blockIdx.x;   // 0..MM-1
  const int b   = bi / NN;
  const int tid = threadIdx.x;
  __shared__ int   sAdj[KK];
  __shared__ float sAtt[KK * CC];
  __shared__ float sSoft[KK * 16];   // [k][c], cols 10..15 kept zero
  __shared__ float sWhT[KK * 16];    // [k][s], cols 12..15 zero (padded wh)
  __shared__ float sOut[CC * SOO], red[256];

  if (tid < KK) sAdj[tid] = adj[bi * KK + tid];
  if (tid < KK * 16) sSoft[tid] = 0.0f;
  __syncthreads();

  if (tid < KK * CC) {
    const int k = tid / CC, c = tid % CC;
    const int j = b * NN + sAdj[k];
    const float a = ws[WS_AX + bi * PW + c] + ws[WS_AY + j * PW + c];
    sAtt[tid] = a;
    ws[WS_ATT + bi * KK * CC + tid] = a;   // pre-softmax attention for loss kernel
  }
  if (tid < KK * 16) {
    const int k = tid / 16, s = tid % 16;
    sWhT[tid] = ws[WS_WH + (b * NN + sAdj[k]) * PW + s];   // padding cols are 0
  }
  __syncthreads();

  // attention.mean partial
  red[tid] = (tid < KK * CC) ? sAtt[tid] : 0.0f;
  __syncthreads();
  for (int off = 128; off > 0; off >>= 1) { if (tid < off) red[tid] += red[tid + off]; __syncthreads(); }
  if (tid == 0) atomicAdd(&ws[WS_ACC + 1], red[0]);

  // softmax over K per channel c
  if (tid < CC) {
    float m = -1e30f;
    for (int k = 0; k < KK; ++k) m = fmaxf(m, sAtt[k * CC + tid]);
    float sum = 0.0f;
    float e[KK];
    for (int k = 0; k < KK; ++k) { e[k] = __expf(sAtt[k * CC + tid] - m); sum += e[k]; }
    const float inv = 1.0f / sum;
    for (int k = 0; k < KK; ++k) sSoft[k * 16 + tid] = e[k] * inv;
  }
  __syncthreads();

  // einsum via WMMA: D[c][s] = sum_k soft[k][c] * whT[k][s]; K=20 -> 5 chunks
  if (tid < 32) {
    const int m  = tid & 15;   // A: M index, B: N index (per-lane)
    const int hi = tid >> 4;
    v8f D = {};
    for (int c = 0; c < 5; ++c) {
      const int k0 = c * 4 + (hi << 1);
      v2f A, Bv;
      A.x  = sSoft[k0 * 16 + m];        A.y  = sSoft[(k0 + 1) * 16 + m];
      Bv.x = sWhT[k0 * 16 + m];         Bv.y = sWhT[(k0 + 1) * 16 + m];
      D = __builtin_amdgcn_wmma_f32_16x16x4_f32(false, A, false, Bv, (short)0, D, false, false);
    }
    for (int r = 0; r < 8; ++r) {
      const int cRow = hi * 8 + r;
      if (cRow < CC && m < SOO) {
        const float v = D[r];
        sOut[cRow * SOO + m] = v;
        out[bi * CC * SOO + cRow * SOO + m] = v;
      }
    }
  }
  __syncthreads();

  // cluster_center_loss partial: sum_{c1<c2} max(5 - ||o_c1 - o_c2||^2, 0)^2
  float v = 0.0f;
  if (tid < CC * CC) {
    const int c1 = tid / CC, c2 = tid % CC;
    if (c1 < c2) {
      float d = 0.0f;
      for (int s = 0; s < SOO; ++s) { const float t = sOut[c1 * SOO + s] - sOut[c2 * SOO + s]; d += t * t; }
      const float r = fmaxf(5.0f - d, 0.0f);
      v = r * r;
    }
  }
  red[tid] = v; __syncthreads();
  for (int off = 128; off > 0; off >>= 1) { if (tid < off) red[tid] += red[tid + off]; __syncthreads(); }
  if (tid == 0) atomicAdd(&ws[WS_ACC + 4], red[0]);
}

// ---------------------------------------------------------------------------
// Kernel 3: per (b,i): wct -> h -> feat (leaky MLP), dist_mat, pairwise
//           prob softmax over C, cluster_loss partial, dist sum.
// ---------------------------------------------------------------------------
__global__ void k_mlp(const int* __restrict__ adj,
                      const float* __restrict__ d1w, const float* __restrict__ d1b,
                      const float* __restrict__ d2w, const float* __restrict__ d2b,
                      float* __restrict__ ws) {
  const int bi  = blockIdx.x;
  const int b   = bi / NN;
  const int tid = threadIdx.x;
  __shared__ int   sAdj[KK];
  __shared__ float sWhI[SOO], sWhT[KK * SOO], sH[KK * 12], sFeat[KK * 6];
  __shared__ float sDist[KK * KK], sAttL[KK * CC], sNum[CC], sDen[CC], red[256];

  if (tid < KK)  sAdj[tid] = adj[bi * KK + tid];
  if (tid < SOO) sWhI[tid] = ws[WS_WH + bi * PW + tid];
  if (tid < CC)  { sNum[tid] = 0.0f; sDen[tid] = 0.0f; }
  __syncthreads();
  if (tid < KK * SOO) {
    const int k = tid / SOO, s = tid % SOO;
    sWhT[tid] = ws[WS_WH + (b * NN + sAdj[k]) * PW + s];
  }
  if (tid < KK * CC) sAttL[tid] = ws[WS_ATT + bi * KK * CC + tid];
  __syncthreads();

  // h = leaky( [wh_i | wh_topk] @ d1_w^T + d1_b ), K x 12
  if (tid < KK * 12) {
    const int k = tid / 12, o = tid % 12;
    float acc = d1b[o];
    for (int s = 0; s < SOO; ++s) acc += d1w[o * 24 + s] * sWhI[s];
    for (int s = 0; s < SOO; ++s) acc += d1w[o * 24 + 12 + s] * sWhT[k * SOO + s];
    sH[tid] = (acc > 0.0f) ? acc : NEG_SLOPE * acc;
  }
  __syncthreads();
  // feat = leaky( h @ d2_w^T + d2_b ), K x 6
  if (tid < KK * 6) {
    const int k = tid / 6, j = tid % 6;
    float acc = d2b[j];
    for (int c = 0; c < 12; ++c) acc += d2w[j * 12 + c] * sH[k * 12 + c];
    sFeat[tid] = (acc > 0.0f) ? acc : NEG_SLOPE * acc;
  }
  __syncthreads();

  // dist_mat (full K x K) + its sum
  float dsum = 0.0f;
  for (int p = tid; p < KK * KK; p += 256) {
    const int k1 = p / KK, k2 = p % KK;
    float d = 0.0f;
    for (int j = 0; j < 6; ++j) { const float t = sFeat[k1 * 6 + j] - sFeat[k2 * 6 + j]; d += t * t; }
    sDist[p] = d; dsum += d;
  }
  red[tid] = dsum; __syncthreads();
  for (int off = 128; off > 0; off >>= 1) { if (tid < off) red[tid] += red[tid + off]; __syncthreads(); }
  if (tid == 0) atomicAdd(&ws[WS_ACC + 2], red[0]);
  __syncthreads();

  // prob = softmax_c(att[k1,c]*att[k2,c]) over masked pairs k1<k2
  for (int p = tid; p < KK * KK; p += 256) {
    const int k1 = p / KK, k2 = p % KK;
    if (k1 < k2) {
      float t[CC]; float m = -1e30f;
      for (int c = 0; c < CC; ++c) { t[c] = sAttL[k1 * CC + c] * sAttL[k2 * CC + c]; m = fmaxf(m, t[c]); }
      float sum = 0.0f;
      for (int c = 0; c < CC; ++c) { t[c] = __expf(t[c] - m); sum += t[c]; }
      const float inv = 1.0f / sum;
      const float d = sDist[p];
      for (int c = 0; c < CC; ++c) {
        const float pr = t[c] * inv;
        atomicAdd(&sNum[c], d * pr);
        atomicAdd(&sDen[c], fmaxf(pr, 1e-5f));
      }
    }
  }
  __syncthreads();
  if (tid == 0) {
    const float maskedFloor = (float)(KK * KK - (KK * (KK - 1)) / 2) * 1e-5f;  // 210e-5
    float loss = 0.0f;
    for (int c = 0; c < CC; ++c) loss += sNum[c] / (sDen[c] + maskedFloor);
    atomicAdd(&ws[WS_ACC + 3], loss);
  }
}

// ---------------------------------------------------------------------------
// Kernel 4: finalize the five scalar outputs
// ---------------------------------------------------------------------------
__global__ void k_final(const float* __restrict__ ws, float* __restrict__ out) {
  if (threadIdx.x == 0 && blockIdx.x == 0) {
    const int P = MM * CC * SOO;                          // 794880
    out[P + 0] = ws[WS_ACC + 3] / (float)MM;              // cluster_loss
    out[P + 1] = ws[WS_ACC + 4] / (float)MM;              // cluster_center_loss
    out[P + 2] = ws[WS_ACC + 0] / (float)(MM * SOO);      // wh.mean
    out[P + 3] = ws[WS_ACC + 1] / (float)(MM * KK * CC);  // attention.mean
    out[P + 4] = ws[WS_ACC + 2] / (float)(MM * KK * KK);  // dist_mat.mean
  }
}

// ---------------------------------------------------------------------------
// Launch
// ---------------------------------------------------------------------------
extern "C" void kernel_launch(void* const* d_in, const int* in_sizes, int n_in,
                              void* d_out, int out_size, void* d_ws, size_t ws_size,
                              hipStream_t stream) {
  (void)in_sizes; (void)n_in; (void)out_size; (void)ws_size;
  // 0: fushed_features (unused), 1: input_data, 2: adj_idx, 3: w_weight,
  // 4: a_weight, 5: d1_w, 6: d1_b, 7: d2_w, 8: d2_b
  const float* x   = (const float*)d_in[1];
  const int*   adj = (const int*)  d_in[2];
  const float* ww  = (const float*)d_in[3];
  const float* aw  = (const float*)d_in[4];
  const float* d1w = (const float*)d_in[5];
  const float* d1b = (const float*)d_in[6];
  const float* d2w = (const float*)d_in[7];
  const float* d2b = (const float*)d_in[8];
  float* out = (float*)d_out;
  float* ws  = (float*)d_ws;

  k_setup<<<1, 256, 0, stream>>>(aw, ww, ws);
  const int waves  = (NTILES + TPW - 1) / TPW;       // 138
  const int blocks = (waves + 7) / 8;                // 18
  k_wh_wmma<<<blocks, 256, 0, stream>>>(x, ws);
  k_attn<<<MM, 256, 0, stream>>>(adj, ws, out);
  k_mlp<<<MM, 256, 0, stream>>>(adj, d1w, d1b, d2w, d2b, ws);
  k_final<<<1, 32, 0, stream>>>(ws, out);
}